// NormalizingFlowModel_47631187312975
// MI455X (gfx1250) — compile-verified
//
#include <hip/hip_runtime.h>

typedef __attribute__((ext_vector_type(2))) float v2f;
typedef __attribute__((ext_vector_type(4))) float v4f;
typedef __attribute__((ext_vector_type(8))) float v8f;

#define NL 15
#define SLOPE 0.01f

// LDS sizes (dwords)
#define AFRAG_DW (NL*3*4*64)   // 11520 : A-fragments, column-permuted block-diag weights
#define BIAS_DW  (NL*4*16)     //   960 : concatenated biases B0..B3 (s-net | b-net)
#define W0_DW    (NL*16)       //   240
#define W4_DW    (NL*16)       //   240
#define B4_DW    (NL*2)        //    30

__global__ __launch_bounds__(256) void flow_kernel(
    const float* __restrict__ z,
    const float* __restrict__ W0, const float* __restrict__ B0,
    const float* __restrict__ W1, const float* __restrict__ B1,
    const float* __restrict__ W2, const float* __restrict__ B2,
    const float* __restrict__ W3, const float* __restrict__ B3,
    const float* __restrict__ W4, const float* __restrict__ B4,
    const int*   __restrict__ swaps,
    float* __restrict__ out, int batch)
{
    __shared__ float s_afrag[AFRAG_DW];
    __shared__ float s_bias[BIAS_DW];
    __shared__ float s_w0[W0_DW];
    __shared__ float s_w4[W4_DW];
    __shared__ float s_b4[B4_DW];
    __shared__ int   s_swap[16];

    const int tid = threadIdx.x;

    // ---- Build A-fragments in LDS: A[m, k_local] per (layer i, hidden j, chunk c).
    // Element at (lane, dword d) = Wblk[m = lane&15, f = pi(4c + d + 2*(lane>>4))]
    // with column permutation pi(kappa): f = 2c + (kappa&1) + 8*(kappa>>1),
    // so the raw WMMA D VGPR pairs of the previous layer are valid B fragments.
    for (int e = tid; e < AFRAG_DW; e += blockDim.x) {
        int d    = e & 1;
        int lane = (e >> 1) & 31;
        int c    = (e >> 6) & 3;
        int ij   = e >> 8;            // i*3 + j
        int i = ij / 3, j = ij - 3*i;
        int m  = lane & 15;
        int kl = d + 2*(lane >> 4);   // K_local 0..3
        int f  = 2*c + (kl & 1) + 8*(kl >> 1);
        const float* Wp = (j == 0) ? W1 : ((j == 1) ? W2 : W3);
        float v = 0.0f;
        if (m < 8 && f < 8)          v = Wp[(i*2+0)*64 + m*8 + f];
        else if (m >= 8 && f >= 8)   v = Wp[(i*2+1)*64 + (m-8)*8 + (f-8)];
        s_afrag[e] = v;
    }
    // Concatenated biases: feature f<8 -> s-net, f>=8 -> b-net
    for (int e = tid; e < BIAS_DW; e += blockDim.x) {
        int f = e & 15; int j = (e >> 4) & 3; int i = e >> 6;
        int g = f >> 3, ff = f & 7;
        const float* Bp = (j == 0) ? B0 : ((j == 1) ? B1 : ((j == 2) ? B2 : B3));
        s_bias[e] = Bp[(i*2+g)*8 + ff];
    }
    for (int e = tid; e < W0_DW; e += blockDim.x) {
        int f = e & 15; int i = e >> 4; int g = f >> 3, ff = f & 7;
        s_w0[e] = W0[(i*2+g)*8 + ff];   // W0: (L,2,8,1)
        s_w4[e] = W4[(i*2+g)*8 + ff];   // W4: (L,2,1,8)
    }
    for (int e = tid; e < B4_DW; e += blockDim.x) s_b4[e] = B4[e];
    for (int e = tid; e < 14;    e += blockDim.x) s_swap[e] = swaps[e];
    __syncthreads();

    const int lane = tid & 31;
    const int hf   = lane >> 4;   // 0: s-net features / x-output, 1: b-net features / logdet
    const int sl   = lane & 15;   // sample-in-tile (replicated across halves)
    const int wid    = (blockIdx.x * blockDim.x + tid) >> 5;
    const int nwaves = (gridDim.x * blockDim.x) >> 5;
    const int ntiles = batch >> 4;

    for (int tile = wid; tile < ntiles; tile += nwaves) {
        const int smp = tile * 16 + sl;
        v2f zz = *(const v2f*)(z + 2*smp);
        float xI = zz.x, xII = zz.y;
        float logdet = 0.0f;

        for (int i = 0; i < NL; ++i) {
            // ---- input layer (rank-1): h0[f] = lrelu(xI * W0cat[f] + B0cat[f])
            float h[8];
            v4f w0a = *(const v4f*)&s_w0[i*16 + 8*hf];
            v4f w0b = *(const v4f*)&s_w0[i*16 + 8*hf + 4];
            v4f b0a = *(const v4f*)&s_bias[(i*4+0)*16 + 8*hf];
            v4f b0b = *(const v4f*)&s_bias[(i*4+0)*16 + 8*hf + 4];
            #pragma unroll
            for (int r = 0; r < 4; ++r) { float t = fmaf(xI, w0a[r], b0a[r]); h[r]   = fmaxf(t, SLOPE*t); }
            #pragma unroll
            for (int r = 0; r < 4; ++r) { float t = fmaf(xI, w0b[r], b0b[r]); h[4+r] = fmaxf(t, SLOPE*t); }

            // ---- hidden layers: D = Wblk(perm) x H + bias via 4 chained f32 WMMAs
            #pragma unroll
            for (int j = 1; j <= 3; ++j) {
                v4f ba = *(const v4f*)&s_bias[(i*4+j)*16 + 8*hf];
                v4f bb = *(const v4f*)&s_bias[(i*4+j)*16 + 8*hf + 4];
                v8f acc;
                acc[0]=ba[0]; acc[1]=ba[1]; acc[2]=ba[2]; acc[3]=ba[3];
                acc[4]=bb[0]; acc[5]=bb[1]; acc[6]=bb[2]; acc[7]=bb[3];
                const float* fr = &s_afrag[(i*3 + (j-1)) * 4 * 64];
                #pragma unroll
                for (int c = 0; c < 4; ++c) {
                    v2f a = *(const v2f*)&fr[c*64 + lane*2];   // ds_load_b64
                    v2f b; b.x = h[2*c]; b.y = h[2*c+1];       // raw D pair == B fragment
                    acc = __builtin_amdgcn_wmma_f32_16x16x4_f32(
                        false, a, false, b, (short)0, acc, false, false);
                }
                #pragma unroll
                for (int r = 0; r < 8; ++r) { float t = acc[r]; h[r] = fmaxf(t, SLOPE*t); }
            }

            // ---- output layer: low half -> log_s[sample], high half -> b[sample]
            float o = s_b4[i*2 + hf];
            v4f w4a = *(const v4f*)&s_w4[i*16 + 8*hf];
            v4f w4b = *(const v4f*)&s_w4[i*16 + 8*hf + 4];
            #pragma unroll
            for (int r = 0; r < 4; ++r) o = fmaf(h[r],   w4a[r], o);
            #pragma unroll
            for (int r = 0; r < 4; ++r) o = fmaf(h[4+r], w4b[r], o);

            float other = __shfl_xor(o, 16, 32);        // pair log_s with b per sample
            float ls  = hf ? other : o;
            float bb2 = hf ? o : other;
            logdet += ls;
            float y = fmaf(__expf(ls), xII, bb2);

            if (i < NL-1 && s_swap[i] > 0) { xII = xI; xI = y; }
            else                           { xII = y; }
        }

        if (hf == 0) {                                   // lanes 0-15: x pairs
            v2f xo; xo.x = xI; xo.y = xII;
            *(v2f*)(out + 2*smp) = xo;
        } else {                                         // lanes 16-31: log_det
            out[2*batch + smp] = logdet;
        }
    }
}

extern "C" void kernel_launch(void* const* d_in, const int* in_sizes, int n_in,
                              void* d_out, int out_size, void* d_ws, size_t ws_size,
                              hipStream_t stream) {
    const int batch = in_sizes[0] / 2;
    flow_kernel<<<dim3(256), dim3(256), 0, stream>>>(
        (const float*)d_in[0],
        (const float*)d_in[1],  (const float*)d_in[2],
        (const float*)d_in[3],  (const float*)d_in[4],
        (const float*)d_in[5],  (const float*)d_in[6],
        (const float*)d_in[7],  (const float*)d_in[8],
        (const float*)d_in[9],  (const float*)d_in[10],
        (const int*)d_in[11],
        (float*)d_out, batch);
}